// _ECELoss_43593918054448
// MI455X (gfx1250) — compile-verified
//
#include <hip/hip_runtime.h>
#include <hip/hip_bf16.h>
#include <math.h>

#define N_BINS 15
#define NCLS   100
#define WAVES_PER_BLOCK 8
#define THREADS (WAVES_PER_BLOCK * 32)
#define MAX_BLOCKS 2048
#define ROW_B128 ((NCLS * 4) / 16)   // 25 x b128 (16B) transfers per 400B row

// 16-byte vector type matching the async-LDS builtin's pointer parameter type
typedef int v4i __attribute__((__vector_size__(4 * sizeof(int))));
typedef __attribute__((address_space(1))) v4i* gv4i_ptr;
typedef __attribute__((address_space(3))) v4i* lv4i_ptr;

// ---- gfx1250 async global->LDS path (guarded so compile never hard-fails) ----
#if defined(__has_builtin)
#  if __has_builtin(__builtin_amdgcn_global_load_async_to_lds_b128)
#    define HAVE_ASYNC_LDS 1
#  endif
#  if __has_builtin(__builtin_amdgcn_s_wait_asynccnt)
#    define WAIT_ASYNC(n) __builtin_amdgcn_s_wait_asynccnt(n)
#  endif
#endif
#ifndef WAIT_ASYNC
#  define WAIT_ASYNC(n) asm volatile("s_wait_asynccnt %0" :: "i"(n) : "memory")
#endif

// Per-row reduction: v0..v3 are this lane's elements (indices lane, lane+32,
// lane+64, lane+96; v3 = -inf for lanes >= 4). Wave32 cross-lane reduce.
__device__ __forceinline__ void accum_row(float v0, float v1, float v2, float v3,
                                          int lane, int row,
                                          const int* __restrict__ labels,
                                          float* bacc)
{
    // max + argmax (lowest index on ties, matching jnp.argmax)
    float m = v0; int mi = lane;
    if (v1 > m) { m = v1; mi = lane + 32; }
    if (v2 > m) { m = v2; mi = lane + 64; }
    if (v3 > m) { m = v3; mi = lane + 96; }
#pragma unroll
    for (int off = 16; off >= 1; off >>= 1) {
        float om = __shfl_xor(m, off, 32);
        int   oi = __shfl_xor(mi, off, 32);
        if (om > m || (om == m && oi < mi)) { m = om; mi = oi; }
    }
    // sum of exp(l - max); top-1 prob = 1/sum
    float s = expf(v0 - m) + expf(v1 - m) + expf(v2 - m);
    if (lane < 4) s += expf(v3 - m);
#pragma unroll
    for (int off = 16; off >= 1; off >>= 1)
        s += __shfl_xor(s, off, 32);

    if (lane == 0) {
        float conf = 1.0f / s;
        int bin = (int)ceilf(conf * (float)N_BINS) - 1;
        bin = bin < 0 ? 0 : (bin > (N_BINS - 1) ? (N_BINS - 1) : bin);
        float hit = (mi == labels[row]) ? 1.0f : 0.0f;
        atomicAdd(&bacc[bin],              1.0f);   // counts
        atomicAdd(&bacc[N_BINS + bin],     conf);   // conf_sum
        atomicAdd(&bacc[2 * N_BINS + bin], hit);    // acc_sum
    }
}

__global__ void ece_zero45(double* g)
{
    int t = threadIdx.x;
    if (t < 3 * N_BINS) g[t] = 0.0;
}

__global__ void __launch_bounds__(THREADS)
ece_pass1(const float* __restrict__ logits, const int* __restrict__ labels,
          double* __restrict__ gacc, int N)
{
    __shared__ __align__(16) float bufs[WAVES_PER_BLOCK][2][NCLS];
    __shared__ float bacc[3 * N_BINS];

    const int tid  = threadIdx.x;
    const int lane = tid & 31;
    const int wave = tid >> 5;
    if (tid < 3 * N_BINS) bacc[tid] = 0.0f;
    __syncthreads();

    const int stride = gridDim.x * WAVES_PER_BLOCK;
    int row = blockIdx.x * WAVES_PER_BLOCK + wave;

#ifdef HAVE_ASYNC_LDS
    // Double-buffered async global->LDS staging: one wave streams its own rows.
    int cur = 0;
    if (row < N && lane < ROW_B128) {
        float* gp = const_cast<float*>(logits) + (size_t)row * NCLS;
        __builtin_amdgcn_global_load_async_to_lds_b128(
            (gv4i_ptr)(gp + lane * 4),
            (lv4i_ptr)(&bufs[wave][0][lane * 4]),
            0, 0);
    }
    for (; row < N; row += stride) {
        int nrow = row + stride;
        if (nrow < N) {
            if (lane < ROW_B128) {
                float* gp = const_cast<float*>(logits) + (size_t)nrow * NCLS;
                __builtin_amdgcn_global_load_async_to_lds_b128(
                    (gv4i_ptr)(gp + lane * 4),
                    (lv4i_ptr)(&bufs[wave][cur ^ 1][lane * 4]),
                    0, 0);
            }
            WAIT_ASYNC(1);   // current buffer's transfer (issued earlier) is done
        } else {
            WAIT_ASYNC(0);
        }
        const float* rp = bufs[wave][cur];
        float v0 = rp[lane];
        float v1 = rp[lane + 32];
        float v2 = rp[lane + 64];
        float v3 = (lane < 4) ? rp[lane + 96] : -__builtin_inff();
        accum_row(v0, v1, v2, v3, lane, row, labels, bacc);
        cur ^= 1;
    }
#else
    // Fallback: direct coalesced global loads.
    for (; row < N; row += stride) {
        const float* g = logits + (size_t)row * NCLS;
        float v0 = g[lane];
        float v1 = g[lane + 32];
        float v2 = g[lane + 64];
        float v3 = (lane < 4) ? g[lane + 96] : -__builtin_inff();
        accum_row(v0, v1, v2, v3, lane, row, labels, bacc);
    }
#endif

    __syncthreads();
    if (tid < 3 * N_BINS)
        atomicAdd(&gacc[tid], (double)bacc[tid]);  // global_atomic_add_f64
}

__global__ void ece_finalize(const double* __restrict__ gacc,
                             float* __restrict__ out, int N)
{
    int t = threadIdx.x;
    double per = 0.0;
    if (t < N_BINS) {
        double cnt = gacc[t];
        double cs  = gacc[N_BINS + t];
        double as  = gacc[2 * N_BINS + t];
        if (cnt > 0.0)
            per = fabs(cs / cnt - as / cnt) * (cnt / (double)N);
    }
#pragma unroll
    for (int off = 16; off >= 1; off >>= 1)
        per += __shfl_xor(per, off, 32);
    if (t == 0) out[0] = (float)per;
}

extern "C" void kernel_launch(void* const* d_in, const int* in_sizes, int n_in,
                              void* d_out, int out_size, void* d_ws, size_t ws_size,
                              hipStream_t stream)
{
    const float* logits = (const float*)d_in[0];
    const int*   labels = (const int*)d_in[1];   // labels (integer) per harness convention
    const int N = in_sizes[1];                   // one label per row

    double* gacc = (double*)d_ws;                // 45 doubles of scratch

    ece_zero45<<<1, 64, 0, stream>>>(gacc);

    int blocks = (N + WAVES_PER_BLOCK - 1) / WAVES_PER_BLOCK;
    if (blocks > MAX_BLOCKS) blocks = MAX_BLOCKS;
    if (blocks < 1) blocks = 1;
    ece_pass1<<<blocks, THREADS, 0, stream>>>(logits, labels, gacc, N);

    ece_finalize<<<1, 32, 0, stream>>>(gacc, (float*)d_out, N);
}